// EquivarLayerTorch_5196910428400
// MI455X (gfx1250) — compile-verified
//
#include <hip/hip_runtime.h>

#define N_ATOMS 50000
#define N_PAIRS 800000
#define CCH     64

typedef __attribute__((ext_vector_type(2))) float v2f;
typedef __attribute__((ext_vector_type(8))) float v8f;

// ---------------- Phase 0: zero the 38.4MB accumulator (float4 stores) ----
__global__ __launch_bounds__(256) void zero4_kernel(float4* __restrict__ p, int n4) {
    int t = blockIdx.x * blockDim.x + threadIdx.x;
    if (t < n4) p[t] = make_float4(0.f, 0.f, 0.f, 0.f);
}

// ---------------- Phase 1: gather + gate + atomic scatter-add -------------
// 4 pairs per 256-thread block; 64 channels (2 waves) per pair.
// 3 non-returning global f32 atomics per thread into the L2-resident agg.
__global__ __launch_bounds__(256) void scatter_kernel(
    const int*   __restrict__ ind2,   // (N_PAIRS, 2)
    const float* __restrict__ p3,     // (N_ATOMS, 3, 64)
    const float* __restrict__ i1,     // (N_PAIRS, 64)
    const float* __restrict__ d3,     // (N_PAIRS, 3)
    float*       __restrict__ agg)    // (N_ATOMS, 3, 64)
{
    const int pl = threadIdx.x >> 6;          // pair within block: 0..3
    const int c  = threadIdx.x & 63;          // channel
    const int p  = blockIdx.x * 4 + pl;       // grid sized exactly: always valid

    const int i = ind2[2 * p];
    const int j = ind2[2 * p + 1];

    // Prefetch the streaming gate a few pairs ahead (speculative, OOB-safe).
    __builtin_prefetch((const void*)(i1 + (size_t)(p + 64) * CCH + c), 0, 1);

    const float gate = i1[(size_t)p * CCH + c];
    const float dx0  = d3[3 * p + 0];
    const float dx1  = d3[3 * p + 1];
    const float dx2  = d3[3 * p + 2];

    const float* src = p3 + (size_t)j * (3 * CCH) + c;   // random gather, L2-hit
    const float  v0  = (src[0]         + dx0) * gate;
    const float  v1  = (src[CCH]       + dx1) * gate;
    const float  v2  = (src[2 * CCH]   + dx2) * gate;

    float* dst = agg + (size_t)i * (3 * CCH) + c;
    (void)__hip_atomic_fetch_add(dst,            v0, __ATOMIC_RELAXED, __HIP_MEMORY_SCOPE_AGENT);
    (void)__hip_atomic_fetch_add(dst + CCH,      v1, __ATOMIC_RELAXED, __HIP_MEMORY_SCOPE_AGENT);
    (void)__hip_atomic_fetch_add(dst + 2 * CCH,  v2, __ATOMIC_RELAXED, __HIP_MEMORY_SCOPE_AGENT);
}

// ---------------- Phase 2: fp32 WMMA GEMM  D = A x W^T --------------------
// A: (150000, 64) = p3_agg rows (atom,x), K = 64 channels.
// B[k][n] = W[n][k]  (W row-major (d, c)).
// One wave per 16x16 output tile; 4 waves/block cover all 64 output cols.
// 16 chained V_WMMA_F32_16X16X4_F32 per tile (K = 16 * 4).
__global__ __launch_bounds__(128) void gemm_wmma_kernel(
    const float* __restrict__ A,      // (150000, 64)
    const float* __restrict__ W,      // (64, 64)
    float*       __restrict__ D)      // (150000, 64)
{
    const int wave = threadIdx.x >> 5;        // n-tile 0..3
    const int lane = threadIdx.x & 31;
    const int tm   = blockIdx.x;              // m-tile 0..9374

    const int m16  = lane & 15;               // M (A) / N (B,C,D) within tile
    const int koff = (lane >> 4) * 2;         // K sub-offset: lanes 16-31 hold K=2,3

    const float* arow = A + (size_t)(tm * 16 + m16) * CCH + koff;   // A[M][K..]
    const float* wrow = W + (size_t)(wave * 16 + m16) * CCH + koff; // W[N][K..] = B[K..][N]

    v8f acc = {};
    #pragma unroll
    for (int k = 0; k < 16; ++k) {
        v2f a = *(const v2f*)(arow + k * 4);
        v2f b = *(const v2f*)(wrow + k * 4);
        acc = __builtin_amdgcn_wmma_f32_16x16x4_f32(
            /*neg_a=*/false, a, /*neg_b=*/false, b,
            /*c_mod=*/(short)0, acc, /*reuse_a=*/false, /*reuse_b=*/false);
    }

    // C/D layout: lane L -> col = L&15; rows 8*(L>>4) + r (r = VGPR index 0..7)
    const int col     = wave * 16 + m16;
    const int rowbase = tm * 16 + (lane >> 4) * 8;
    float* o = D + (size_t)rowbase * CCH + col;
    #pragma unroll
    for (int r = 0; r < 8; ++r)
        o[(size_t)r * CCH] = acc[r];
}

// ---------------- Phase 3: dotted[a][d] = sum_x p3_new[a][x][d]^2 ---------
__global__ __launch_bounds__(256) void dotted_kernel(
    const float* __restrict__ p3n,    // (N_ATOMS, 3, 64)
    float*       __restrict__ dotted) // (N_ATOMS, 64)
{
    int t = blockIdx.x * blockDim.x + threadIdx.x;
    if (t >= N_ATOMS * CCH) return;
    int a = t >> 6, d = t & 63;
    const float* p = p3n + (size_t)a * (3 * CCH) + d;
    float x0 = p[0], x1 = p[CCH], x2 = p[2 * CCH];
    dotted[t] = x0 * x0 + x1 * x1 + x2 * x2;
}

extern "C" void kernel_launch(void* const* d_in, const int* in_sizes, int n_in,
                              void* d_out, int out_size, void* d_ws, size_t ws_size,
                              hipStream_t stream) {
    const int*   ind2 = (const int*)  d_in[0];   // (800000, 2) int32
    const float* p3   = (const float*)d_in[1];   // (50000, 3, 64)
    const float* i1   = (const float*)d_in[2];   // (800000, 64)
    const float* d3   = (const float*)d_in[3];   // (800000, 3)
    const float* W    = (const float*)d_in[4];   // (64, 64)

    float* p3_new = (float*)d_out;                               // 9.6M f32
    float* dotted = (float*)d_out + (size_t)N_ATOMS * 3 * CCH;   // 3.2M f32
    float* agg    = (float*)d_ws;                                // 9.6M f32 = 38.4MB

    const int nagg  = N_ATOMS * 3 * CCH;        // 9,600,000 (divisible by 4)
    const int nagg4 = nagg / 4;

    zero4_kernel<<<(nagg4 + 255) / 256, 256, 0, stream>>>((float4*)agg, nagg4);

    scatter_kernel<<<N_PAIRS / 4, 256, 0, stream>>>(ind2, p3, i1, d3, agg);

    gemm_wmma_kernel<<<(N_ATOMS * 3) / 16, 128, 0, stream>>>(agg, W, p3_new);

    dotted_kernel<<<(N_ATOMS * CCH + 255) / 256, 256, 0, stream>>>(p3_new, dotted);
}